// GCN_67937792688660
// MI455X (gfx1250) — compile-verified
//
#include <hip/hip_runtime.h>

#define D 128
#define NEG_SLOPE 0.1f

typedef float v2f __attribute__((ext_vector_type(2)));
typedef float v8f __attribute__((ext_vector_type(8)));

__device__ __forceinline__ float lrelu(float x) { return x >= 0.0f ? x : NEG_SLOPE * x; }

__device__ __forceinline__ v8f wmma_f32_step(v2f a, v2f b, v8f c) {
  return __builtin_amdgcn_wmma_f32_16x16x4_f32(
      /*neg_a=*/false, a, /*neg_b=*/false, b,
      /*c_mod=*/(short)0, c, /*reuse_a=*/false, /*reuse_b=*/false);
}

// ---------------------------------------------------------------------------
// Tiled fp32 WMMA GEMM:  C[M x NC] = op(A[M x K]) * B[K x NC]
// Block = 256 threads (8 waves). Each block owns a 16-row strip of A, staged
// once into LDS with coalesced float4 loads (optional fused leaky-ReLU).
// Wave w computes the 16x16 output tile at columns w*16 using
// V_WMMA_F32_16X16X4_F32, K-loop fully unrolled; K and NC are compile-time so
// all B-load / C-store offsets fold into instruction immediates.
// Requires: M % 16 == 0, NC == 16 * (blockDim.x/32), K % 4 == 0.
// ---------------------------------------------------------------------------
template <bool LRELU_A, int K, int NC>
__global__ void gemm_wmma_tiled(const float* __restrict__ A,
                                const float* __restrict__ B,
                                float* __restrict__ Cmat) {
  constexpr int PAD = 2;                 // row pad -> conflict-free ds_load_b64
  __shared__ float As[16][K + PAD];

  const int tid  = threadIdx.x;
  const int row0 = blockIdx.x * 16;

  // Cooperative, coalesced stage of the 16 x K A-strip into LDS.
  constexpr int VEC = 4;
  constexpr int TOT = 16 * K / VEC;      // float4 elements
  const float* Ablk = A + (size_t)row0 * K;
  for (int i = tid; i < TOT; i += 256) {
    const int r = i / (K / VEC);
    const int c = (i % (K / VEC)) * VEC;
    float4 v = *(const float4*)(Ablk + (size_t)r * K + c);
    if (LRELU_A) {
      v.x = lrelu(v.x); v.y = lrelu(v.y); v.z = lrelu(v.z); v.w = lrelu(v.w);
    }
    As[r][c + 0] = v.x; As[r][c + 1] = v.y;
    As[r][c + 2] = v.z; As[r][c + 3] = v.w;
  }
  __syncthreads();

  const int wave = tid >> 5;
  const int lane = tid & 31;
  const int half = lane >> 4;
  const int l15  = lane & 15;
  const int col0 = wave * 16;

  const float* Bcol = B + col0 + l15;    // all further offsets compile-time

  v8f acc = {};
#pragma unroll
  for (int k0 = 0; k0 < K; k0 += 4) {
    const int ka = k0 + 2 * half;
    const v2f a = *(const v2f*)&As[l15][ka];   // ds_load_b64, banks 2*l15(+1)
    v2f b;
    b.x = Bcol[ka * NC];                       // imm-offset global_load_b32
    b.y = Bcol[ka * NC + NC];
    acc = wmma_f32_step(a, b, acc);
  }

  float* Crow = Cmat + (size_t)(row0 + 8 * half) * NC + col0 + l15;
#pragma unroll
  for (int r = 0; r < 8; ++r) {
    Crow[r * NC] = acc[r];                     // imm-offset stores
  }
}

// One wave per 16x16 tile (tiny classifier GEMM). Exact tiles only.
template <int K, int NC>
__global__ void gemm_wmma_small(const float* __restrict__ A,
                                const float* __restrict__ B,
                                const float* __restrict__ bias,
                                float* __restrict__ Cmat) {
  const int lane = threadIdx.x & 31;
  const int half = lane >> 4;
  const int l15  = lane & 15;
  const int row0 = blockIdx.x * 16;
  const int col0 = blockIdx.y * 16;

  const float* Arow = A + (size_t)(row0 + l15) * K;
  const float* Bcol = B + col0 + l15;

  v8f acc = {};
#pragma unroll
  for (int k0 = 0; k0 < K; k0 += 4) {
    const int ka = k0 + 2 * half;
    const v2f a = *(const v2f*)(Arow + ka);    // contiguous pair -> b64 load
    v2f b;
    b.x = Bcol[ka * NC];
    b.y = Bcol[ka * NC + NC];
    acc = wmma_f32_step(a, b, acc);
  }

  const float bs = bias[col0 + l15];
  float* Crow = Cmat + (size_t)(row0 + 8 * half) * NC + col0 + l15;
#pragma unroll
  for (int r = 0; r < 8; ++r) {
    Crow[r * NC] = acc[r] + bs;
  }
}

// ---------------------------------------------------------------------------
// Elementwise / scatter kernels
// ---------------------------------------------------------------------------
__global__ void k_init(float* dinv, int N, float* gsum, float* cnt, int G) {
  const int i = blockIdx.x * blockDim.x + threadIdx.x;
  if (i < N) dinv[i] = 1.0f;               // self-loop contribution to degree
  if (i < G * D) gsum[i] = 0.0f;
  if (i < G) cnt[i] = 0.0f;
}

__global__ void k_deg(const int* __restrict__ dst, float* deg, int E) {
  const int e = blockIdx.x * blockDim.x + threadIdx.x;
  if (e < E) atomicAdd(&deg[dst[e]], 1.0f);
}

__global__ void k_rsqrt(float* dinv, int N) {
  const int i = blockIdx.x * blockDim.x + threadIdx.x;
  if (i < N) dinv[i] = rsqrtf(dinv[i]);
}

// agg[n,f] = h[n,f] * dinv[n]^2 + bias[f]   (self-loop term + bias)
__global__ void k_init_agg(const float* __restrict__ h,
                           const float* __restrict__ dinv,
                           const float* __restrict__ bias,
                           float* __restrict__ agg, int N) {
  const int idx = blockIdx.x * blockDim.x + threadIdx.x;
  if (idx >= N * D) return;
  const int n = idx >> 7;
  const int f = idx & (D - 1);
  const float di = dinv[n];
  agg[idx] = h[idx] * di * di + bias[f];
}

// One wave per edge: agg[dst] += dinv[src]*dinv[dst] * h[src]  (128 feats)
__global__ void k_scatter(const float* __restrict__ h,
                          const int* __restrict__ src,
                          const int* __restrict__ dst,
                          const float* __restrict__ dinv,
                          float* __restrict__ agg, int E) {
  const int e = blockIdx.x * (blockDim.x >> 5) + (threadIdx.x >> 5);
  if (e >= E) return;
  const int lane = threadIdx.x & 31;
  const int s = src[e];
  const int d = dst[e];
  const float c = dinv[s] * dinv[d];
  const float* hs = h + (size_t)s * D;
  float* ad = agg + (size_t)d * D;
#pragma unroll
  for (int j = 0; j < D / 32; ++j) {
    const int f = lane + 32 * j;
    atomicAdd(&ad[f], c * hs[f]);
  }
}

// gsum[batch[n], f] += lrelu(agg2[n, f]);  cnt[batch[n]] += 1 (once per node)
__global__ void k_pool(const float* __restrict__ agg2,
                       const int* __restrict__ batch,
                       float* __restrict__ gsum, float* __restrict__ cnt, int N) {
  const int idx = blockIdx.x * blockDim.x + threadIdx.x;
  if (idx >= N * D) return;
  const int n = idx >> 7;
  const int f = idx & (D - 1);
  const int g = batch[n];
  atomicAdd(&gsum[g * D + f], lrelu(agg2[idx]));
  if (f == 0) atomicAdd(&cnt[g], 1.0f);
}

__global__ void k_gmean(const float* __restrict__ gsum,
                        const float* __restrict__ cnt,
                        float* __restrict__ gmean, int G) {
  const int idx = blockIdx.x * blockDim.x + threadIdx.x;
  if (idx >= G * D) return;
  const int g = idx >> 7;
  gmean[idx] = gsum[idx] / fmaxf(cnt[g], 1.0f);
}

// ---------------------------------------------------------------------------
extern "C" void kernel_launch(void* const* d_in, const int* in_sizes, int n_in,
                              void* d_out, int out_size, void* d_ws, size_t ws_size,
                              hipStream_t stream) {
  const float* x     = (const float*)d_in[0];
  const int*   ei    = (const int*)d_in[1];
  const int*   batch = (const int*)d_in[2];
  const float* W1    = (const float*)d_in[3];
  const float* b1    = (const float*)d_in[4];
  const float* W2    = (const float*)d_in[5];
  const float* b2    = (const float*)d_in[6];
  const float* Wl    = (const float*)d_in[7];
  const float* bl    = (const float*)d_in[8];

  const int N = in_sizes[0] / D;       // 50000  (multiple of 16)
  const int E = in_sizes[1] / 2;       // 600000
  const int C = in_sizes[8];           // 16
  const int G = out_size / C;          // 64

  const int* src = ei;
  const int* dst = ei + E;

  // Workspace (floats): bufA[N*D] | bufB[N*D] | dinv[N] | gsum[G*D] | cnt[G] | gmean[G*D]
  float* bufA  = (float*)d_ws;
  float* bufB  = bufA + (size_t)N * D;
  float* dinv  = bufB + (size_t)N * D;
  float* gsum  = dinv + N;
  float* cnt   = gsum + (size_t)G * D;
  float* gmean = cnt + G;

  const int TB = 256;
  const int initMax = (N > G * D) ? N : G * D;
  const int rowStrips = N / 16;        // exact

  // 1) degrees -> dinv
  k_init<<<(initMax + TB - 1) / TB, TB, 0, stream>>>(dinv, N, gsum, cnt, G);
  k_deg<<<(E + TB - 1) / TB, TB, 0, stream>>>(dst, dinv, E);
  k_rsqrt<<<(N + TB - 1) / TB, TB, 0, stream>>>(dinv, N);

  // 2) layer 1: h1 = x @ W1  (LDS-staged WMMA, 8 waves/block)
  gemm_wmma_tiled<false, D, D><<<rowStrips, 256, 0, stream>>>(x, W1, bufA);
  // 3) agg1 = h1*dinv^2 + b1, then scatter edges
  k_init_agg<<<((size_t)N * D + TB - 1) / TB, TB, 0, stream>>>(bufA, dinv, b1, bufB, N);
  k_scatter<<<(E + (TB / 32) - 1) / (TB / 32), TB, 0, stream>>>(bufA, src, dst, dinv, bufB, E);

  // 4) layer 2: h2 = lrelu(agg1) @ W2  (leaky-ReLU fused into LDS stage)
  gemm_wmma_tiled<true, D, D><<<rowStrips, 256, 0, stream>>>(bufB, W2, bufA);
  // 5) agg2 = h2*dinv^2 + b2, then scatter edges
  k_init_agg<<<((size_t)N * D + TB - 1) / TB, TB, 0, stream>>>(bufA, dinv, b2, bufB, N);
  k_scatter<<<(E + (TB / 32) - 1) / (TB / 32), TB, 0, stream>>>(bufA, src, dst, dinv, bufB, E);

  // 6) global mean pool over lrelu(agg2)
  k_pool<<<((size_t)N * D + TB - 1) / TB, TB, 0, stream>>>(bufB, batch, gsum, cnt, N);
  k_gmean<<<(G * D + TB - 1) / TB, TB, 0, stream>>>(gsum, cnt, gmean, G);

  // 7) classifier: out = gmean @ Wl + bl   (WMMA, 4x1 tiles of 16x16)
  dim3 finGrid(G / 16, C / 16);
  gemm_wmma_small<D, 16><<<finGrid, 32, 0, stream>>>(gmean, Wl, bl, (float*)d_out);
}